// LightGCN_83897891160077
// MI455X (gfx1250) — compile-verified
//
#include <hip/hip_runtime.h>

#define N_USERS 100000
#define N_ITEMS 50000
#define N_NODES 150000
#define DIM 256
#define NNZ_E 12800000
#define CHUNK 32   // edges staged per async chunk (one b32 per lane)

#if defined(__has_builtin)
#if __has_builtin(__builtin_amdgcn_global_load_async_to_lds_b32)
#define LG_ASYNC 1
#endif
#endif

#ifdef LG_ASYNC
typedef __attribute__((address_space(1))) int g_int;   // global AS pointer elem
typedef __attribute__((address_space(3))) int l_int;   // LDS AS pointer elem
#endif

// ---------------------------------------------------------------------------
// Kernel 1: build X = concat(user, item), out = X (acc init), zero histogram
// ---------------------------------------------------------------------------
__global__ void lg_init(const float* __restrict__ u, const float* __restrict__ it,
                        float* __restrict__ X, float* __restrict__ out,
                        unsigned* __restrict__ counts) {
    const long long total4 = (long long)N_NODES * DIM / 4;  // 9,600,000 float4
    const long long user4  = (long long)N_USERS * DIM / 4;  // 6,400,000 float4
    const long long stride = (long long)gridDim.x * blockDim.x;
    long long i0 = (long long)blockIdx.x * blockDim.x + threadIdx.x;
    for (long long i = i0; i < total4; i += stride) {
        const float4 v = (i < user4) ? ((const float4*)u)[i]
                                     : ((const float4*)it)[i - user4];
        ((float4*)X)[i]   = v;
        ((float4*)out)[i] = v;
    }
    for (long long i = i0; i < N_NODES; i += stride) counts[i] = 0u;
}

// ---------------------------------------------------------------------------
// Kernel 2: histogram of edge rows
// ---------------------------------------------------------------------------
__global__ void lg_hist(const int* __restrict__ rows, unsigned* __restrict__ counts) {
    const int e = blockIdx.x * blockDim.x + threadIdx.x;
    if (e < NNZ_E) atomicAdd(&counts[rows[e]], 1u);
}

// ---------------------------------------------------------------------------
// Kernel 3: single-block exclusive scan: counts -> row_ptr (counts becomes
// the scatter cursor, initialized to the row start offsets)
// ---------------------------------------------------------------------------
__global__ void __launch_bounds__(1024) lg_scan(unsigned* __restrict__ counts,
                                                int* __restrict__ row_ptr) {
    __shared__ unsigned s[1024];
    const int t = threadIdx.x;
    const int chunk = (N_NODES + 1023) / 1024;  // 147
    const int lo = t * chunk;
    const int hi = (lo + chunk < N_NODES) ? (lo + chunk) : N_NODES;
    unsigned sum = 0;
    for (int i = lo; i < hi; ++i) sum += counts[i];
    s[t] = sum;
    __syncthreads();
    for (int off = 1; off < 1024; off <<= 1) {
        unsigned v = (t >= off) ? s[t - off] : 0u;
        __syncthreads();
        s[t] += v;
        __syncthreads();
    }
    unsigned run = s[t] - sum;  // exclusive prefix of this thread's chunk
    for (int i = lo; i < hi; ++i) {
        unsigned c = counts[i];
        row_ptr[i] = (int)run;
        counts[i]  = run;
        run += c;
    }
    if (t == 1023) row_ptr[N_NODES] = (int)s[1023];
}

// ---------------------------------------------------------------------------
// Kernel 4: scatter edges into CSR order
// ---------------------------------------------------------------------------
__global__ void lg_scatter(const int* __restrict__ rows, const int* __restrict__ cols,
                           const float* __restrict__ vals, unsigned* __restrict__ cursor,
                           int* __restrict__ csr_col, float* __restrict__ csr_val) {
    const int e = blockIdx.x * blockDim.x + threadIdx.x;
    if (e < NNZ_E) {
        const unsigned p = atomicAdd(&cursor[rows[e]], 1u);
        csr_col[p] = cols[e];
        csr_val[p] = vals[e];
    }
}

// ---------------------------------------------------------------------------
// Kernel 5: CSR SpMM, one wave32 per row, fused accumulator update.
// Edge metadata is streamed global->LDS with the gfx1250 async copy engine,
// double-buffered (32-edge chunks), consumed via ds_load_b64 broadcasts.
//   last==0: Y[row] = sum; out[row] += sum
//   last==1: out[row] = (out[row] + sum) * 0.25f
// ---------------------------------------------------------------------------
__global__ void __launch_bounds__(256) lg_spmm(const int* __restrict__ row_ptr,
                                               const int* __restrict__ csr_col,
                                               const float* __restrict__ csr_val,
                                               const float* __restrict__ X,
                                               float* __restrict__ Y,
                                               float* __restrict__ out,
                                               int last) {
    const int lane  = threadIdx.x & 31;
    const int wslot = threadIdx.x >> 5;
    const int row   = blockIdx.x * 8 + wslot;   // 18750*8 == 150000, no tail
    if (row >= N_NODES) return;

    const int start = __builtin_amdgcn_readfirstlane(row_ptr[row]);
    const int end   = __builtin_amdgcn_readfirstlane(row_ptr[row + 1]);

    float4 a0 = {0.f, 0.f, 0.f, 0.f};
    float4 a1 = {0.f, 0.f, 0.f, 0.f};

#ifdef LG_ASYNC
    // (col,val) interleaved per edge, per wave, double buffered: 4KB/block
    __shared__ __align__(16) int sPair[8][2][CHUNK * 2];

    const int cnt = end - start;
    const int nch = (cnt + CHUNK - 1) / CHUNK;

    // Issue async copy of chunk ch into buffer (ch&1): one b32 per lane from
    // csr_col and csr_val, landing interleaved as (col,val) pairs in LDS.
    #define LG_ISSUE(ch)                                                          \
        do {                                                                      \
            int e_ = start + (ch) * CHUNK + lane;                                 \
            e_ = (e_ < NNZ_E) ? e_ : (NNZ_E - 1);                                 \
            int* dc_ = &sPair[wslot][(ch) & 1][lane * 2];                         \
            __builtin_amdgcn_global_load_async_to_lds_b32(                        \
                (g_int*)(csr_col + e_), (l_int*)dc_, 0, 0);                       \
            __builtin_amdgcn_global_load_async_to_lds_b32(                        \
                (g_int*)(csr_val + e_), (l_int*)(dc_ + 1), 0, 0);                 \
        } while (0)

    if (nch > 0) LG_ISSUE(0);
    for (int ch = 0; ch < nch; ++ch) {
        if (ch + 1 < nch) {
            LG_ISSUE(ch + 1);
            asm volatile("s_wait_asynccnt 0x2" ::: "memory");  // chunk ch landed
        } else {
            asm volatile("s_wait_asynccnt 0x0" ::: "memory");
        }
        const int2* pb = (const int2*)&sPair[wslot][ch & 1][0];
        const int n = (cnt - ch * CHUNK < CHUNK) ? (cnt - ch * CHUNK) : CHUNK;
        for (int j = 0; j < n; ++j) {
            const int2 cv = pb[j];  // uniform-address ds_load_b64 broadcast
            const int   c = __builtin_amdgcn_readfirstlane(cv.x);
            const float v = __int_as_float(__builtin_amdgcn_readfirstlane(cv.y));
            const float4* xr = (const float4*)(X + (size_t)c * DIM);
            const float4 x0 = xr[lane];
            const float4 x1 = xr[lane + 32];
            a0.x = fmaf(v, x0.x, a0.x); a0.y = fmaf(v, x0.y, a0.y);
            a0.z = fmaf(v, x0.z, a0.z); a0.w = fmaf(v, x0.w, a0.w);
            a1.x = fmaf(v, x1.x, a1.x); a1.y = fmaf(v, x1.y, a1.y);
            a1.z = fmaf(v, x1.z, a1.z); a1.w = fmaf(v, x1.w, a1.w);
        }
        // All ds reads of this buffer must retire before the async engine may
        // overwrite it next iteration (WAR on LDS).
        asm volatile("s_wait_dscnt 0x0" ::: "memory");
    }
    #undef LG_ISSUE
#else
    for (int j = start; j < end; ++j) {
        const int   c = __builtin_amdgcn_readfirstlane(csr_col[j]);
        const float v = __int_as_float(
            __builtin_amdgcn_readfirstlane(__float_as_int(csr_val[j])));
        if (j + 1 < end) {
            const int cn = __builtin_amdgcn_readfirstlane(csr_col[j + 1]);
            __builtin_prefetch(X + (size_t)cn * DIM + (size_t)(lane & 7) * 32, 0, 1);
        }
        const float4* xr = (const float4*)(X + (size_t)c * DIM);
        const float4 x0 = xr[lane];
        const float4 x1 = xr[lane + 32];
        a0.x = fmaf(v, x0.x, a0.x); a0.y = fmaf(v, x0.y, a0.y);
        a0.z = fmaf(v, x0.z, a0.z); a0.w = fmaf(v, x0.w, a0.w);
        a1.x = fmaf(v, x1.x, a1.x); a1.y = fmaf(v, x1.y, a1.y);
        a1.z = fmaf(v, x1.z, a1.z); a1.w = fmaf(v, x1.w, a1.w);
    }
#endif

    const size_t base = (size_t)row * DIM;
    float4* orow = (float4*)(out + base);
    float4 o0 = orow[lane];
    float4 o1 = orow[lane + 32];

    if (last) {
        o0.x = (o0.x + a0.x) * 0.25f; o0.y = (o0.y + a0.y) * 0.25f;
        o0.z = (o0.z + a0.z) * 0.25f; o0.w = (o0.w + a0.w) * 0.25f;
        o1.x = (o1.x + a1.x) * 0.25f; o1.y = (o1.y + a1.y) * 0.25f;
        o1.z = (o1.z + a1.z) * 0.25f; o1.w = (o1.w + a1.w) * 0.25f;
    } else {
        float4* yr = (float4*)(Y + base);
        yr[lane]      = a0;
        yr[lane + 32] = a1;
        o0.x += a0.x; o0.y += a0.y; o0.z += a0.z; o0.w += a0.w;
        o1.x += a1.x; o1.y += a1.y; o1.z += a1.z; o1.w += a1.w;
    }
    orow[lane]      = o0;
    orow[lane + 32] = o1;
}

// ---------------------------------------------------------------------------
extern "C" void kernel_launch(void* const* d_in, const int* in_sizes, int n_in,
                              void* d_out, int out_size, void* d_ws, size_t ws_size,
                              hipStream_t stream) {
    const float* user_emb  = (const float*)d_in[0];
    const float* item_emb  = (const float*)d_in[1];
    const int*   edge_rows = (const int*)d_in[2];
    const int*   edge_cols = (const int*)d_in[3];
    const float* edge_vals = (const float*)d_in[4];
    float* out = (float*)d_out;

    char* ws = (char*)d_ws;
    const size_t embBytes = (size_t)N_NODES * DIM * sizeof(float);  // 153,600,000
    float*    bufA    = (float*)(ws);
    float*    bufB    = (float*)(ws + embBytes);
    unsigned* counts  = (unsigned*)(ws + 2 * embBytes);                       // 600,000 B
    int*      row_ptr = (int*)(ws + 2 * embBytes + 600000);                   // 600,004 B
    int*      csr_col = (int*)(ws + 2 * embBytes + 1200016);                  // 51.2 MB
    float*    csr_val = (float*)(ws + 2 * embBytes + 1200016 + (size_t)NNZ_E * 4);

    lg_init<<<37500, 256, 0, stream>>>(user_emb, item_emb, bufA, out, counts);
    lg_hist<<<NNZ_E / 256, 256, 0, stream>>>(edge_rows, counts);
    lg_scan<<<1, 1024, 0, stream>>>(counts, row_ptr);
    lg_scatter<<<NNZ_E / 256, 256, 0, stream>>>(edge_rows, edge_cols, edge_vals,
                                                counts, csr_col, csr_val);

    const int spmmBlocks = (N_NODES + 7) / 8;  // 18750, one wave32 per row
    lg_spmm<<<spmmBlocks, 256, 0, stream>>>(row_ptr, csr_col, csr_val, bufA, bufB, out, 0);
    lg_spmm<<<spmmBlocks, 256, 0, stream>>>(row_ptr, csr_col, csr_val, bufB, bufA, out, 0);
    lg_spmm<<<spmmBlocks, 256, 0, stream>>>(row_ptr, csr_col, csr_val, bufA, bufB, out, 1);
}